// GiddLinearNoise_78855599555354
// MI455X (gfx1250) — compile-verified
//
#include <hip/hip_runtime.h>

#ifndef __has_builtin
#define __has_builtin(x) 0
#endif

#define WAVE 32
// VPAD = 393 * 128: multiple of 128 (= 32 lanes * 4 floats), >= 50257
#define VPAD 50304

typedef int v4i __attribute__((ext_vector_type(4)));

__device__ __forceinline__ unsigned mix32(unsigned x) {
  // lowbias32 finalizer: cheap, well-mixed counter-based PRNG
  x ^= x >> 16; x *= 0x7feb352dU;
  x ^= x >> 15; x *= 0x846ca68bU;
  x ^= x >> 16;
  return x;
}

__global__ __launch_bounds__(256) void gidd_sample_zt_kernel(
    const int* __restrict__ ids,   // (B*S,)
    const float* __restrict__ t,   // (B,)
    const float* __restrict__ pi,  // (V,)
    float* __restrict__ out,       // (B*S,)
    int n_pos, int S, int V) {
  // Full pi vector staged in gfx1250's 320KB LDS (201KB needed).
  __shared__ float s_pi[VPAD];

  const int tid  = threadIdx.x;
  const int nthr = blockDim.x;

  // ---- Stage pi: async global -> LDS, 16B per lane per issue ----
  const int n4 = V >> 2;  // number of full float4 chunks (covers 4*n4 floats)
#if __has_builtin(__builtin_amdgcn_global_load_async_to_lds_b128)
  for (int c = tid; c < n4; c += nthr) {
    __builtin_amdgcn_global_load_async_to_lds_b128(
        (__attribute__((address_space(1))) v4i*)(pi + 4 * c),
        (__attribute__((address_space(3))) v4i*)(&s_pi[4 * c]),
        /*offset=*/0, /*cpol=*/0);
  }
#else
  for (int i = tid; i < (n4 << 2); i += nthr) s_pi[i] = pi[i];
#endif
  // tail elements not covered by 16B chunks
  for (int i = (n4 << 2) + tid; i < V; i += nthr) s_pi[i] = pi[i];
  // zero padding: pad entries give prob 0 and can never win the argmax
  for (int i = V + tid; i < VPAD; i += nthr) s_pi[i] = 0.0f;

#if __has_builtin(__builtin_amdgcn_s_wait_asynccnt)
  __builtin_amdgcn_s_wait_asynccnt(0);
#elif defined(__gfx1250__)
  asm volatile("s_wait_asynccnt 0" ::: "memory");
#endif
  __syncthreads();

  // ---- One wave32 per (b, s) position ----
  const int lane = tid & (WAVE - 1);
  const int pos  = (blockIdx.x * nthr + tid) / WAVE;
  if (pos >= n_pos) return;   // whole wave exits together

  const int   x     = ids[pos];
  const float tb    = t[pos / S];
  const float alpha = 1.0f - tb;   // weight on onehot(x)

  float best = -1.0f;  // all real scores are >= 0
  int   bidx = 0;

  const float4*  p4   = (const float4*)s_pi;
  const unsigned seed = (unsigned)pos * (unsigned)VPAD;

  // 393 iterations/lane; ds_load_b128 feeds 4 elements per trip
  for (int i = lane; i < VPAD / 4; i += WAVE) {
    const float4 q  = p4[i];
    const int    v0 = i << 2;
    const float  pv[4] = {q.x, q.y, q.z, q.w};
#pragma unroll
    for (int j = 0; j < 4; ++j) {
      const int   v = v0 + j;
      // probs = t*pi[v] + (v==x ? (1-t) : 0)
      const float p = tb * pv[j] + ((v == x) ? alpha : 0.0f);
      // uniform u in [0,1) from counter hash on (pos, v)
      const unsigned h = mix32(seed + (unsigned)v);
      const float u = (float)(h >> 8) * (1.0f / 16777216.0f);
      // gumbel_norm = 1e-10 - log(u + 1e-10)  (always > 0)
      const float g = 1e-10f - __logf(u + 1e-10f);
      const float s = p * __builtin_amdgcn_rcpf(g);
      if (s > best) { best = s; bidx = v; }
    }
  }

  // ---- wave32 argmax reduction (ds_bpermute shuffles) ----
#pragma unroll
  for (int off = WAVE / 2; off > 0; off >>= 1) {
    const float os = __shfl_xor(best, off, WAVE);
    const int   oi = __shfl_xor(bidx, off, WAVE);
    if (os > best || (os == best && oi < bidx)) { best = os; bidx = oi; }
  }

  if (lane == 0) out[pos] = (float)bidx;
}

extern "C" void kernel_launch(void* const* d_in, const int* in_sizes, int n_in,
                              void* d_out, int out_size, void* d_ws, size_t ws_size,
                              hipStream_t stream) {
  (void)n_in; (void)d_ws; (void)ws_size; (void)out_size;
  const int*   ids = (const int*)d_in[0];    // input_ids (B,S) int32
  const float* t   = (const float*)d_in[1];  // t (B,) f32
  const float* pi  = (const float*)d_in[2];  // pi (V,) f32
  float*       out = (float*)d_out;

  const int n_pos = in_sizes[0];             // B*S = 4096
  const int B     = in_sizes[1];             // 4
  const int S     = n_pos / (B > 0 ? B : 1); // 1024
  const int V     = in_sizes[2];             // 50257

  const int threads = 256;                       // 8 wave32s / block
  const int waves_per_block = threads / WAVE;
  const int blocks = (n_pos + waves_per_block - 1) / waves_per_block;

  gidd_sample_zt_kernel<<<blocks, threads, 0, stream>>>(ids, t, pi, out,
                                                        n_pos, S, V);
}